// DeformableAttention_1039382086382
// MI455X (gfx1250) — compile-verified
//
#include <hip/hip_runtime.h>

typedef __attribute__((ext_vector_type(2))) float v2f;
typedef __attribute__((ext_vector_type(8))) float v8f;

#define B_   4
#define C_   64
#define H_   256
#define W_   256
#define HW_  65536          // H_*W_
#define R_   4
#define PPB  128            // pixels per block in kernel 1
#define WSTR 65             // padded LDS row stride for weights (bank-conflict-free A frags)

// ---------------------------------------------------------------------------
// Kernel 0: fuse Wo (8x64) @ Wq (64x64) -> WOQ (padded to 16x64), and bias.
// off = Wo*(Wq*x + bq) + bo  ==>  off = (Wo*Wq)*x + (Wo*bq + bo)
// ---------------------------------------------------------------------------
__global__ void fuse_wo_wq(const float* __restrict__ Wo, const float* __restrict__ Wq,
                           const float* __restrict__ bq, const float* __restrict__ bo,
                           float* __restrict__ WOQ, float* __restrict__ BOQ) {
  int tid = threadIdx.x;
  if (tid < 16 * 64) {
    int r = tid >> 6, c = tid & 63;
    float s = 0.f;
    if (r < 8) {
      for (int o = 0; o < 64; ++o) s += Wo[r * 64 + o] * Wq[o * 64 + c];
    }
    WOQ[tid] = s;                      // rows 8..15 are zero padding
  }
  if (tid < 16) {
    float s = 0.f;
    if (tid < 8) {
      for (int o = 0; o < 64; ++o) s += Wo[tid * 64 + o] * bq[o];
      s += bo[tid];
    }
    BOQ[tid] = s;
  }
}

// ---------------------------------------------------------------------------
// Kernel 1: fused QKV+offset projection GEMM using V_WMMA_F32_16X16X4_F32.
// Combined weight matrix: rows 0-63 Wq | 64-127 Wk | 128-191 Wv | 192-207 Woq(pad)
// => 13 M-tiles of 16 rows. Each wave computes a 16(pixel) column strip for all
// 13 tiles: 16 K-steps x 13 wmma, accumulators seeded with the biases.
// ---------------------------------------------------------------------------
__global__ void __launch_bounds__(256)
qkv_proj(const float* __restrict__ x,
         const float* __restrict__ Wq, const float* __restrict__ bq,
         const float* __restrict__ Wk, const float* __restrict__ bk,
         const float* __restrict__ Wv, const float* __restrict__ bv,
         const float* __restrict__ WOQ, const float* __restrict__ BOQ,
         float* __restrict__ Q, float* __restrict__ K, float* __restrict__ V,
         float* __restrict__ OFF) {
  __shared__ float sW[208 * WSTR];       // 54,080 B
  __shared__ float sB[208];              //    832 B
  __shared__ float sX[C_ * PPB];         // 32,768 B   (total ~88 KB of 320 KB)

  const int tid = threadIdx.x;

  // Stage weights with padded stride (A-frag reads become conflict-lite).
  for (int i = tid; i < 64 * 64; i += 256) {
    int r = i >> 6, c = i & 63;
    sW[r * WSTR + c]         = Wq[i];
    sW[(64 + r) * WSTR + c]  = Wk[i];
    sW[(128 + r) * WSTR + c] = Wv[i];
  }
  for (int i = tid; i < 16 * 64; i += 256) {
    int r = i >> 6, c = i & 63;
    sW[(192 + r) * WSTR + c] = WOQ[i];
  }
  for (int i = tid; i < 208; i += 256) {
    sB[i] = (i < 64) ? bq[i] : (i < 128) ? bk[i - 64]
          : (i < 192) ? bv[i - 128] : BOQ[i - 192];
  }

  // Stage x tile: 64 channels x 128 pixels, coalesced 512B row reads.
  const long long P = (long long)blockIdx.x * PPB;   // linear over B*HW
  const int b     = (int)(P >> 16);
  const int pbase = (int)(P & (HW_ - 1));
  const float* xb = x + (long long)b * C_ * HW_;
  for (int i = tid; i < C_ * PPB; i += 256) {
    int c = i >> 7, col = i & (PPB - 1);
    sX[i] = xb[c * HW_ + pbase + col];
  }
  __syncthreads();

  const int lane = tid & 31;
  const int wave = tid >> 5;
  const int half = lane >> 4;        // 0: K 0/1 (A) & N rows, 1: K 2/3
  const int n    = lane & 15;
  const int colLocal = wave * 16 + n;

  // Accumulators seeded with bias: D-tile layout VGPR j -> row j (+8 upper half).
  v8f acc[13];
#pragma unroll
  for (int mt = 0; mt < 13; ++mt) {
#pragma unroll
    for (int j = 0; j < 8; ++j) acc[mt][j] = sB[mt * 16 + half * 8 + j];
  }

#pragma unroll 2
  for (int k = 0; k < 16; ++k) {
    const int kb = k * 4 + half * 2;
    v2f bfrag;                                    // B 4x16: N across lanes 0-15
    bfrag[0] = sX[(kb + 0) * PPB + colLocal];
    bfrag[1] = sX[(kb + 1) * PPB + colLocal];
#pragma unroll
    for (int mt = 0; mt < 13; ++mt) {
      v2f afrag;                                  // A 16x4: M across lanes 0-15
      const int row = mt * 16 + n;
      afrag[0] = sW[row * WSTR + kb + 0];
      afrag[1] = sW[row * WSTR + kb + 1];
      acc[mt] = __builtin_amdgcn_wmma_f32_16x16x4_f32(
          /*neg_a=*/false, afrag, /*neg_b=*/false, bfrag,
          /*c_mod=*/(short)0, acc[mt], /*reuse_a=*/false, /*reuse_b=*/false);
    }
  }

  // Store Q/K/V in (B,C,HW) layout; off rows 192..199 only (200..207 = pad).
  const long long chw = (long long)C_ * HW_;
  const int col = pbase + colLocal;
  float* Qb = Q + (long long)b * chw;
  float* Kb = K + (long long)b * chw;
  float* Vb = V + (long long)b * chw;
  float* Ob = OFF + (long long)b * 8 * HW_;
#pragma unroll
  for (int mt = 0; mt < 13; ++mt) {
#pragma unroll
    for (int j = 0; j < 8; ++j) {
      const int row = mt * 16 + half * 8 + j;
      const float v = acc[mt][j];
      if (row < 64)        Qb[row * HW_ + col] = v;
      else if (row < 128)  Kb[(row - 64) * HW_ + col] = v;
      else if (row < 192)  Vb[(row - 128) * HW_ + col] = v;
      else if (row < 200)  Ob[(row - 192) * HW_ + col] = v;
    }
  }
}

// ---------------------------------------------------------------------------
// Kernel 2: per-pixel deformable attention. One lane per pixel; offsets are
// near-zero so gathers are near-coalesced; K/V/Q mostly resident in 192MB L2.
// ---------------------------------------------------------------------------
__global__ void __launch_bounds__(256)
deform_attend(const float* __restrict__ Q, const float* __restrict__ K,
              const float* __restrict__ V, const float* __restrict__ OFF,
              float* __restrict__ out) {
  const long long t = (long long)blockIdx.x * blockDim.x + threadIdx.x;
  const int b = (int)(t >> 16);
  const int p = (int)(t & (HW_ - 1));
  const int py = p >> 8, px = p & (W_ - 1);

  const long long chw = (long long)C_ * HW_;
  const float* Qb = Q + (long long)b * chw;
  const float* Kb = K + (long long)b * chw;
  const float* Vb = V + (long long)b * chw;
  const float* Ob = OFF + (long long)b * 8 * HW_;

  int idx[R_];
#pragma unroll
  for (int r = 0; r < R_; ++r) {
    const float ox = Ob[(2 * r) * HW_ + p];
    const float oy = Ob[(2 * r + 1) * HW_ + p];
    const float fx = fminf(fmaxf((float)px + ox, 0.f), (float)(W_ - 1));
    const float fy = fminf(fmaxf((float)py + oy, 0.f), (float)(H_ - 1));
    idx[r] = ((int)fy) * W_ + (int)fx;   // trunc of clamped non-negative == floor
  }

  float l0 = 0.f, l1 = 0.f, l2 = 0.f, l3 = 0.f;
#pragma unroll 4
  for (int c = 0; c < C_; ++c) {
    const float* Kc = Kb + (long long)c * HW_;
    const float q = Qb[(long long)c * HW_ + p];
    l0 += q * Kc[idx[0]];
    l1 += q * Kc[idx[1]];
    l2 += q * Kc[idx[2]];
    l3 += q * Kc[idx[3]];
  }
  const float scale = 0.125f;            // 1/sqrt(64)
  const float a0 = 1.f / (1.f + __expf(-l0 * scale));
  const float a1 = 1.f / (1.f + __expf(-l1 * scale));
  const float a2 = 1.f / (1.f + __expf(-l2 * scale));
  const float a3 = 1.f / (1.f + __expf(-l3 * scale));

  float* ob = out + (long long)b * chw;
#pragma unroll 4
  for (int c = 0; c < C_; ++c) {
    const float* Vc = Vb + (long long)c * HW_;
    ob[(long long)c * HW_ + p] =
        a0 * Vc[idx[0]] + a1 * Vc[idx[1]] + a2 * Vc[idx[2]] + a3 * Vc[idx[3]];
  }
}

// ---------------------------------------------------------------------------
extern "C" void kernel_launch(void* const* d_in, const int* in_sizes, int n_in,
                              void* d_out, int out_size, void* d_ws, size_t ws_size,
                              hipStream_t stream) {
  const float* x  = (const float*)d_in[0];
  const float* Wq = (const float*)d_in[1];
  const float* bq = (const float*)d_in[2];
  const float* Wk = (const float*)d_in[3];
  const float* bk = (const float*)d_in[4];
  const float* Wv = (const float*)d_in[5];
  const float* bv = (const float*)d_in[6];
  const float* Wo = (const float*)d_in[7];
  const float* bo = (const float*)d_in[8];

  float* ws = (float*)d_ws;
  const long long NQ = (long long)B_ * C_ * HW_;        // 16,777,216 floats
  float* Q   = ws;
  float* K   = ws + NQ;
  float* V   = ws + 2 * NQ;
  float* OFF = ws + 3 * NQ;                              // B * 8 * HW
  float* WOQ = OFF + (long long)B_ * 8 * HW_;            // 16*64 padded
  float* BOQ = WOQ + 16 * 64;

  float* out = (float*)d_out;

  fuse_wo_wq<<<1, 1024, 0, stream>>>(Wo, Wq, bq, bo, WOQ, BOQ);

  dim3 g1((B_ * HW_) / PPB);                             // 2048 blocks
  qkv_proj<<<g1, 256, 0, stream>>>(x, Wq, bq, Wk, bk, Wv, bv, WOQ, BOQ,
                                   Q, K, V, OFF);

  dim3 g2((B_ * HW_) / 256);                             // 1024 blocks
  deform_attend<<<g2, 256, 0, stream>>>(Q, K, V, OFF, out);
}